// ListMLELoss_13271448944950
// MI455X (gfx1250) — compile-verified
//
#include <hip/hip_runtime.h>
#include <cstdint>

#define NEGC  (-1.0e9f)
#define ROWK  256
#define INF_BITS 0x7F800000u

typedef float v2f __attribute__((ext_vector_type(2)));
typedef float v8f __attribute__((ext_vector_type(8)));

// ---------------------------------------------------------------------------
// Kernel 1: one workgroup (256 threads = 8 wave32) per row.
//  - async global->LDS stage of scores/ranks (CDNA5 ASYNCcnt path)
//  - bitonic sort of (rank_bits, idx) u64 keys  == stable argsort by rank
//  - suffix logsumexp scan (Hillis-Steele, associative (m, w) combine)
//  - masked term sum -> rowloss[row]
// ---------------------------------------------------------------------------
__global__ __launch_bounds__(256) void listmle_row_kernel(
    const float* __restrict__ scores,
    const float* __restrict__ ranks,
    const unsigned char* __restrict__ mask,
    float* __restrict__ rowloss)
{
    __shared__ float scoreLds[ROWK];
    __shared__ float rankLds[ROWK];
    __shared__ unsigned long long keyLds[ROWK];
    __shared__ float mArr[ROWK];
    __shared__ float wArr[ROWK];
    __shared__ float red[ROWK];

    const int tid = threadIdx.x;
    const int row = blockIdx.x;
    const long long base = (long long)row * ROWK + tid;

    // ---- CDNA5 async copy: global -> LDS, tracked by ASYNCcnt -------------
    {
        unsigned long long gS = (unsigned long long)(uintptr_t)(scores + base);
        unsigned long long gR = (unsigned long long)(uintptr_t)(ranks  + base);
        // low 32 bits of a generic pointer to LDS == LDS byte offset
        unsigned int ldsS = (unsigned int)(uintptr_t)&scoreLds[tid];
        unsigned int ldsR = (unsigned int)(uintptr_t)&rankLds[tid];
        asm volatile("global_load_async_to_lds_b32 %0, %1, off"
                     :: "v"(ldsS), "v"(gS) : "memory");
        asm volatile("global_load_async_to_lds_b32 %0, %1, off"
                     :: "v"(ldsR), "v"(gR) : "memory");
        asm volatile("s_wait_asynccnt 0" ::: "memory");
    }
    const unsigned char mk = mask[base];   // bool -> 1 byte
    __syncthreads();

    // ---- build sortable key: (masked rank bits << 32) | original index ----
    {
        unsigned int rb = mk ? __float_as_uint(rankLds[tid]) : INF_BITS;
        keyLds[tid] = ((unsigned long long)rb << 32) | (unsigned int)tid;
    }
    __syncthreads();

    // ---- bitonic sort, ascending, 256 elements / 256 threads --------------
    for (int k = 2; k <= ROWK; k <<= 1) {
        for (int j = k >> 1; j > 0; j >>= 1) {
            int ix = tid ^ j;
            if (ix > tid) {
                unsigned long long a = keyLds[tid];
                unsigned long long b = keyLds[ix];
                bool asc = ((tid & k) == 0);
                if ((a > b) == asc) { keyLds[tid] = b; keyLds[ix] = a; }
            }
            __syncthreads();
        }
    }

    // ---- gather sorted score, recover mask --------------------------------
    unsigned long long key = keyLds[tid];
    unsigned int idx = (unsigned int)key;
    unsigned int rb  = (unsigned int)(key >> 32);
    bool valid = (rb < INF_BITS);
    float ss = valid ? scoreLds[idx & (ROWK - 1)] : NEGC;   // TEMPERATURE == 1

    // ---- suffix (reverse inclusive) logsumexp scan ------------------------
    float mi = ss, wi = 1.0f;
    mArr[tid] = mi; wArr[tid] = wi;
    __syncthreads();
    for (int d = 1; d < ROWK; d <<= 1) {
        float m2 = 0.0f, w2 = 0.0f;
        bool has = (tid + d) < ROWK;
        if (has) { m2 = mArr[tid + d]; w2 = wArr[tid + d]; }
        __syncthreads();
        if (has) {
            float nm = fmaxf(mi, m2);
            wi = wi * expf(mi - nm) + w2 * expf(m2 - nm);
            mi = nm;
        }
        mArr[tid] = mi; wArr[tid] = wi;
        __syncthreads();
    }
    float lse  = mi + logf(wi);
    float term = valid ? (lse - ss) : 0.0f;

    // ---- block reduction (deterministic fixed tree) -----------------------
    red[tid] = term;
    __syncthreads();
    if (tid < 128) red[tid] += red[tid + 128];
    __syncthreads();
    if (tid < 64)  red[tid] += red[tid + 64];
    __syncthreads();
    if (tid < 32) {
        float v = red[tid] + red[tid + 32];
        for (int o = 16; o > 0; o >>= 1) v += __shfl_down(v, o, 32);
        if (tid == 0) rowloss[row] = v;
    }
}

// ---------------------------------------------------------------------------
// Kernel 2: deterministic reduction of 32768 row losses -> mean.
// Final 64-way stage uses V_WMMA_F32_16X16X4_F32 with B = ones:
//   D[m][n] = sum_k A[m][k], so per-lane sums of D give the total.
// ---------------------------------------------------------------------------
__global__ __launch_bounds__(256) void listmle_reduce_kernel(
    const float* __restrict__ rowloss, float* __restrict__ out, int n)
{
    __shared__ float red[256];
    const int tid = threadIdx.x;

    float acc = 0.0f;
    for (int i = tid; i < n; i += 256) acc += rowloss[i];
    red[tid] = acc;
    __syncthreads();
    if (tid < 128) red[tid] += red[tid + 128];
    __syncthreads();
    if (tid < 64)  red[tid] += red[tid + 64];
    __syncthreads();
    // red[0..63] hold 64 partials. Every wave (uniform control flow, EXEC all
    // ones) runs the same WMMA reduction; only thread 0 writes.
    int lane = tid & 31;
    int m  = lane & 15;
    int k0 = (lane < 16) ? 0 : 2;           // f32 16x4 A layout (2 VGPRs)
    v2f a; a.x = red[m * 4 + k0]; a.y = red[m * 4 + k0 + 1];
    v2f b; b.x = 1.0f; b.y = 1.0f;          // ones: B layout irrelevant
    v8f c = {};
    c = __builtin_amdgcn_wmma_f32_16x16x4_f32(
            false, a, false, b, (short)0, c, false, false);
    float s8 = c[0] + c[1] + c[2] + c[3] + c[4] + c[5] + c[6] + c[7];
    // lanes 0-15 hold sum over m=0..7, lanes 16-31 over m=8..15
    float other = __shfl_down(s8, 16, 32);
    float tot = s8 + other;
    if (tid == 0) out[0] = tot / (float)n;
}

// ---------------------------------------------------------------------------
extern "C" void kernel_launch(void* const* d_in, const int* in_sizes, int n_in,
                              void* d_out, int out_size, void* d_ws, size_t ws_size,
                              hipStream_t stream)
{
    (void)n_in; (void)out_size; (void)ws_size;
    const float*         scores = (const float*)d_in[0];
    const float*         ranks  = (const float*)d_in[1];
    const unsigned char* mask   = (const unsigned char*)d_in[2];  // jax bool = 1 byte
    float* out     = (float*)d_out;
    float* rowloss = (float*)d_ws;       // B floats of scratch

    const int BK = in_sizes[0];          // B * K
    const int B  = BK / ROWK;            // 32768

    listmle_row_kernel<<<B, 256, 0, stream>>>(scores, ranks, mask, rowloss);
    listmle_reduce_kernel<<<1, 256, 0, stream>>>(rowloss, out, B);
}